// SPECTEncoder_28226525069458
// MI455X (gfx1250) — compile-verified
//
#include <hip/hip_runtime.h>
#include <hip/hip_bf16.h>

typedef __attribute__((ext_vector_type(2))) float v2f;
typedef __attribute__((ext_vector_type(8))) float v8f;

#define NGRAPH 64
#define HID    32
#define PDIM   256
#define ODIM   1024

// Order-preserving float <-> uint key (total order incl. negatives) for
// exact atomic min/max on floats.
__device__ __forceinline__ unsigned fkey(float f) {
  unsigned u = __float_as_uint(f);
  return (u & 0x80000000u) ? ~u : (u | 0x80000000u);
}
__device__ __forceinline__ float fdec(unsigned k) {
  unsigned u = (k & 0x80000000u) ? (k & 0x7FFFFFFFu) : ~k;
  return __uint_as_float(u);
}

// ---- Stage A: init deg (=1 for self loop) and pooling accumulators ----
__global__ void k_init(float* __restrict__ deg, unsigned* __restrict__ maxkey,
                       unsigned* __restrict__ minkey, unsigned* __restrict__ cnt, int n) {
  int i = blockIdx.x * blockDim.x + threadIdx.x;
  if (i < n) deg[i] = 1.0f;
  if (i < NGRAPH) { maxkey[i] = 0u; minkey[i] = 0xFFFFFFFFu; cnt[i] = 0u; }
}

// ---- Stage B: in-degree via atomics ----
__global__ void k_deg(const int* __restrict__ dst, float* __restrict__ deg, int E) {
  int i = blockIdx.x * blockDim.x + threadIdx.x;
  if (i < E) atomicAdd(&deg[dst[i]], 1.0f);
}

// ---- Stage C: dinv = rsqrt(deg) (in place), agg = self-loop contribution ----
__global__ void k_dinv(const float* __restrict__ x, float* __restrict__ dinv,
                       float* __restrict__ agg, int n) {
  int i = blockIdx.x * blockDim.x + threadIdx.x;
  if (i < n) {
    float r = rsqrtf(dinv[i]);   // deg >= 1 always (self loop)
    dinv[i] = r;
    agg[i] = x[i] * r * r;
  }
}

// ---- Stage D: edge aggregation (scalar message: 32x less atomic traffic) ----
__global__ void k_edge(const int* __restrict__ src, const int* __restrict__ dst,
                       const float* __restrict__ x, const float* __restrict__ dinv,
                       float* __restrict__ agg, int E) {
  int i = blockIdx.x * blockDim.x + threadIdx.x;
  if (i < E) {
    int s = src[i], d = dst[i];
    atomicAdd(&agg[d], x[s] * dinv[s] * dinv[d]);
  }
}

// ---- Stage E: per-graph min/max of agg + counts ----
__global__ void k_pool(const float* __restrict__ agg, const int* __restrict__ batch,
                       unsigned* __restrict__ maxkey, unsigned* __restrict__ minkey,
                       unsigned* __restrict__ cnt, int n) {
  int i = blockIdx.x * blockDim.x + threadIdx.x;
  if (i < n) {
    int g = batch[i];
    unsigned k = fkey(agg[i]);
    atomicMax(&maxkey[g], k);
    atomicMin(&minkey[g], k);
    atomicAdd(&cnt[g], 1u);
  }
}

// ---- Stage F: whole projection MLP in one workgroup (32 waves), f32 WMMA ----
__global__ void __launch_bounds__(1024)
k_mlp(const unsigned* __restrict__ maxkey, const unsigned* __restrict__ minkey,
      const unsigned* __restrict__ cnt,
      const float* __restrict__ W1, const float* __restrict__ b1,
      const float* __restrict__ Wp1, const float* __restrict__ bp1,
      const float* __restrict__ gamma, const float* __restrict__ beta,
      const float* __restrict__ Wp2, const float* __restrict__ bp2,
      float* __restrict__ p,      // scratch [64*256]
      float* __restrict__ out) {  // d_out [64*1024]
  __shared__ float z[NGRAPH][HID];
  __shared__ float red[1024];
  __shared__ float invn[NGRAPH];

  const int tid  = threadIdx.x;
  const int lane = tid & 31;
  const int wave = tid >> 5;
  const int half = lane >> 4;   // 0: K lower pair, 1: K upper pair
  const int l16  = lane & 15;

  // z[g][k] = relu((W1[k]>=0 ? max_agg : min_agg)*W1[k] + b1[k]); empty graph -> 0
  for (int idx = tid; idx < NGRAPH * HID; idx += 1024) {
    int g = idx >> 5, k = idx & 31;
    float w = W1[k];
    float a = (w >= 0.0f) ? fdec(maxkey[g]) : fdec(minkey[g]);
    float v = fmaf(a, w, b1[k]);
    z[g][k] = (cnt[g] != 0u) ? fmaxf(v, 0.0f) : 0.0f;
  }
  __syncthreads();

  // GEMM1: p = z @ Wp1 + bp1   (64x256, K=32) -> 4x16 tiles, 2 per wave
  for (int t = wave * 2; t < wave * 2 + 2; ++t) {
    int mt = t >> 4, nt = t & 15;
    int row = mt * 16 + l16;
    int col = nt * 16 + l16;
    v8f acc = {};
    for (int kk = 0; kk < 8; ++kk) {
      int k0 = kk * 4 + half * 2;
      v2f a; a.x = z[row][k0];              a.y = z[row][k0 + 1];
      v2f b; b.x = Wp1[k0 * PDIM + col];    b.y = Wp1[(k0 + 1) * PDIM + col];
      acc = __builtin_amdgcn_wmma_f32_16x16x4_f32(false, a, false, b,
                                                  (short)0, acc, false, false);
    }
    int orow = mt * 16 + half * 8;
    for (int i = 0; i < 8; ++i)
      p[(orow + i) * PDIM + col] = acc[i] + bp1[col];
  }
  __syncthreads();

  // BatchNorm (training stats over 64 rows) + ReLU, one column per thread
  if (tid < PDIM) {
    int c = tid;
    float m = 0.0f;
    for (int r = 0; r < NGRAPH; ++r) m += p[r * PDIM + c];
    m *= (1.0f / NGRAPH);
    float v = 0.0f;
    for (int r = 0; r < NGRAPH; ++r) { float d = p[r * PDIM + c] - m; v += d * d; }
    v *= (1.0f / NGRAPH);
    float scale = gamma[c] * rsqrtf(v + 1e-5f);
    float shift = beta[c] - m * scale;
    for (int r = 0; r < NGRAPH; ++r)
      p[r * PDIM + c] = fmaxf(fmaf(p[r * PDIM + c], scale, shift), 0.0f);
  }
  __syncthreads();

  // GEMM2: out = p @ Wp2 + bp2   (64x1024, K=256) -> 4x64 tiles, 8 per wave
  for (int j = 0; j < 8; ++j) {
    int t = wave * 8 + j;
    int mt = t >> 6, nt = t & 63;
    int row = mt * 16 + l16;
    int col = nt * 16 + l16;
    v8f acc = {};
    for (int kk = 0; kk < 64; ++kk) {
      int k0 = kk * 4 + half * 2;
      v2f a; a.x = p[row * PDIM + k0];      a.y = p[row * PDIM + k0 + 1];
      v2f b; b.x = Wp2[k0 * ODIM + col];    b.y = Wp2[(k0 + 1) * ODIM + col];
      acc = __builtin_amdgcn_wmma_f32_16x16x4_f32(false, a, false, b,
                                                  (short)0, acc, false, false);
    }
    int orow = mt * 16 + half * 8;
    for (int i = 0; i < 8; ++i)
      out[(orow + i) * ODIM + col] = acc[i] + bp2[col];
  }
  __syncthreads();

  // Row L2 normalize: 16 partial-sum threads per row, then scale
  {
    int r = tid >> 4, part = tid & 15;
    float s = 0.0f;
    int base = r * ODIM + part * 64;
    for (int c = 0; c < 64; ++c) { float v = out[base + c]; s += v * v; }
    red[tid] = s;
  }
  __syncthreads();
  if (tid < NGRAPH) {
    float t = 0.0f;
    for (int k = 0; k < 16; ++k) t += red[tid * 16 + k];
    invn[tid] = 1.0f / fmaxf(sqrtf(t), 1e-12f);
  }
  __syncthreads();
  for (int idx = tid; idx < NGRAPH * ODIM; idx += 1024)
    out[idx] *= invn[idx >> 10];
}

extern "C" void kernel_launch(void* const* d_in, const int* in_sizes, int n_in,
                              void* d_out, int out_size, void* d_ws, size_t ws_size,
                              hipStream_t stream) {
  const float* x     = (const float*)d_in[0];
  const int*   ei    = (const int*)d_in[1];   // [2, E]: row0=src, row1=dst
  const int*   batch = (const int*)d_in[2];
  const float* W1    = (const float*)d_in[3];
  const float* b1    = (const float*)d_in[4];
  const float* Wp1   = (const float*)d_in[5];
  const float* bp1   = (const float*)d_in[6];
  const float* gamma = (const float*)d_in[7];
  const float* beta  = (const float*)d_in[8];
  const float* Wp2   = (const float*)d_in[9];
  const float* bp2   = (const float*)d_in[10];

  const int N = in_sizes[0];
  const int E = in_sizes[1] / 2;

  float*    ws     = (float*)d_ws;
  float*    deg    = ws;                         // N floats, becomes dinv in place
  float*    agg    = ws + N;                     // N floats
  unsigned* maxkey = (unsigned*)(ws + 2 * (size_t)N);
  unsigned* minkey = maxkey + NGRAPH;
  unsigned* cnt    = minkey + NGRAPH;
  float*    pbuf   = (float*)(cnt + NGRAPH);     // 64*256 floats

  const int TB = 256;
  k_init<<<(N + TB - 1) / TB, TB, 0, stream>>>(deg, maxkey, minkey, cnt, N);
  k_deg <<<(E + TB - 1) / TB, TB, 0, stream>>>(ei + E, deg, E);
  k_dinv<<<(N + TB - 1) / TB, TB, 0, stream>>>(x, deg, agg, N);
  k_edge<<<(E + TB - 1) / TB, TB, 0, stream>>>(ei, ei + E, x, deg, agg, E);
  k_pool<<<(N + TB - 1) / TB, TB, 0, stream>>>(agg, batch, maxkey, minkey, cnt, N);
  k_mlp <<<1, 1024, 0, stream>>>(maxkey, minkey, cnt, W1, b1, Wp1, bp1,
                                 gamma, beta, Wp2, bp2, pbuf, (float*)d_out);
}